// DecoderRNN_54339926229038
// MI455X (gfx1250) — compile-verified
//
#include <hip/hip_runtime.h>
#include <math.h>

#define HSZ 2048
#define OSZ 1024
#define GSZ 8192  // 4*HSZ

typedef __attribute__((ext_vector_type(2))) float v2f;
typedef __attribute__((ext_vector_type(8))) float v8f;

// ---------------------------------------------------------------------------
// Wave-level GEMV tile: 16 rows of W (row-major, K cols) dotted with x (in LDS)
// using V_WMMA_F32_16X16X4_F32.
//   A-frag (16x4 f32): lane l in [0,16) holds W[row][k0], W[row][k0+1];
//                      lane l+16     holds W[row][k0+2], W[row][k0+3].
//   B-frag (4x16 f32, broadcast): B[k][n] = x[k0+k] for all n, so lane l<16
//   holds x[k0],x[k0+1] and lane l+16 holds x[k0+2],x[k0+3] — same offset
//   pattern as A -> one global b64 + one ds b64 per lane per wmma.
// Result: every lane's 8-VGPR accumulator holds y[rbase + (lane>=16?8:0) + j].
// ---------------------------------------------------------------------------
__device__ __forceinline__ v8f wave_gemv(const float* __restrict__ W,
                                         const float* xl,
                                         int row, int K, int lane) {
  const int off = (lane >> 4) << 1;             // 0 for lanes 0-15, 2 for 16-31
  const float* wp = W + (size_t)row * (size_t)K + off;
  const float* xp = xl + off;
  v8f acc = {0.f, 0.f, 0.f, 0.f, 0.f, 0.f, 0.f, 0.f};
  for (int k = 0; k < K; k += 16) {
#pragma unroll
    for (int u = 0; u < 16; u += 4) {
      v2f a = *(const v2f*)(wp + k + u);
      v2f b = *(const v2f*)(xp + k + u);
      acc = __builtin_amdgcn_wmma_f32_16x16x4_f32(false, a, false, b,
                                                  (short)0, acc, false, false);
    }
  }
  return acc;
}

// ---------------------------------------------------------------------------
// Batched recurrent GEMV: z[l][r] = dot(Whh_l[r,:], h_l) + bih_l[r] + bhh_l[r]
// for all 5 layers in parallel (independent of the serial chain).
// grid = 5 layers * 64 row-blocks; block = 256 threads = 8 waves * 16 rows.
// ---------------------------------------------------------------------------
struct Whh5Args {
  const float* W[5];
  const float* h[5];
  const float* b0[5];
  const float* b1[5];
  float* z;  // [5][8192]
};

__global__ __launch_bounds__(256) void k_whh5(Whh5Args p) {
  __shared__ float xl[HSZ];
  const int layer = blockIdx.x >> 6;
  const int rb    = blockIdx.x & 63;
  const float* x = p.h[layer];
  for (int i = threadIdx.x; i < HSZ; i += 256) xl[i] = x[i];
  __syncthreads();

  const int lane  = threadIdx.x & 31;
  const int wave  = threadIdx.x >> 5;
  const int rbase = rb * 128 + wave * 16;
  const int row   = rbase + (lane & 15);

  v8f acc = wave_gemv(p.W[layer], xl, row, HSZ, lane);

  if ((lane & 15) == 0) {
    const int r0 = rbase + (lane >> 4) * 8;
    float* zo = p.z + layer * GSZ;
    const float* b0 = p.b0[layer];
    const float* b1 = p.b1[layer];
#pragma unroll
    for (int j = 0; j < 8; ++j) zo[r0 + j] = acc[j] + b0[r0 + j] + b1[r0 + j];
  }
}

// ---------------------------------------------------------------------------
// gates[r] = dot(Wih[r,:], x) + z[r]     (z already contains both biases)
// grid = 64 blocks (8192 rows / 128 rows per block), K = 1024.
// ---------------------------------------------------------------------------
__global__ __launch_bounds__(256) void k_gemv_add(const float* __restrict__ W,
                                                  const float* __restrict__ x,
                                                  const float* __restrict__ addv,
                                                  float* __restrict__ out,
                                                  int K) {
  __shared__ float xl[HSZ];
  for (int i = threadIdx.x; i < K; i += 256) xl[i] = x[i];
  __syncthreads();

  const int lane  = threadIdx.x & 31;
  const int wave  = threadIdx.x >> 5;
  const int rbase = blockIdx.x * 128 + wave * 16;
  const int row   = rbase + (lane & 15);

  v8f acc = wave_gemv(W, xl, row, K, lane);

  if ((lane & 15) == 0) {
    const int r0 = rbase + (lane >> 4) * 8;
#pragma unroll
    for (int j = 0; j < 8; ++j) out[r0 + j] = acc[j] + addv[r0 + j];
  }
}

// ---------------------------------------------------------------------------
// Shared linear head: y[r] = dot(Wout[r,:], h) + bout[r]; also emits the
// residual input for the next layer: xnext = y + yprev (if requested).
// grid = 8 blocks (1024 rows), K = 2048.
// ---------------------------------------------------------------------------
__global__ __launch_bounds__(256) void k_head(const float* __restrict__ Wout,
                                              const float* __restrict__ hvec,
                                              const float* __restrict__ bout,
                                              float* __restrict__ y,
                                              const float* __restrict__ yprev,
                                              float* __restrict__ xnext) {
  __shared__ float xl[HSZ];
  for (int i = threadIdx.x; i < HSZ; i += 256) xl[i] = hvec[i];
  __syncthreads();

  const int lane  = threadIdx.x & 31;
  const int wave  = threadIdx.x >> 5;
  const int rbase = blockIdx.x * 128 + wave * 16;
  const int row   = rbase + (lane & 15);

  v8f acc = wave_gemv(Wout, xl, row, HSZ, lane);

  if ((lane & 15) == 0) {
    const int r0 = rbase + (lane >> 4) * 8;
#pragma unroll
    for (int j = 0; j < 8; ++j) {
      const int r = r0 + j;
      const float yv = acc[j] + bout[r];
      y[r] = yv;
      if (xnext) xnext[r] = yv + (yprev ? yprev[r] : 0.f);
    }
  }
}

// ---------------------------------------------------------------------------
// Pointwise LSTM cell update. gates layout: [i | f | g | o], each HSZ long.
// grid = 8 blocks * 256 threads = 2048.
// ---------------------------------------------------------------------------
__global__ __launch_bounds__(256) void k_lstm_pointwise(const float* __restrict__ gates,
                                                        const float* __restrict__ cin,
                                                        float* __restrict__ hn,
                                                        float* __restrict__ cn) {
  const int j = blockIdx.x * 256 + threadIdx.x;
  const float ig = gates[j];
  const float fg = gates[HSZ + j];
  const float gg = gates[2 * HSZ + j];
  const float og = gates[3 * HSZ + j];
  const float i = 1.f / (1.f + expf(-ig));
  const float f = 1.f / (1.f + expf(-fg));
  const float g = tanhf(gg);
  const float o = 1.f / (1.f + expf(-og));
  const float cv = f * cin[j] + i * g;
  cn[j] = cv;
  hn[j] = o * tanhf(cv);
}

// ---------------------------------------------------------------------------
// Softmax over 1024 logits, single workgroup.
// ---------------------------------------------------------------------------
__global__ __launch_bounds__(256) void k_softmax(const float* __restrict__ logits,
                                                 float* __restrict__ out) {
  __shared__ float red[256];
  const int t = threadIdx.x;
  float v0 = logits[t], v1 = logits[t + 256], v2 = logits[t + 512], v3 = logits[t + 768];
  float m = fmaxf(fmaxf(v0, v1), fmaxf(v2, v3));
  red[t] = m;
  __syncthreads();
  for (int s = 128; s > 0; s >>= 1) {
    if (t < s) red[t] = fmaxf(red[t], red[t + s]);
    __syncthreads();
  }
  m = red[0];
  __syncthreads();
  v0 = expf(v0 - m); v1 = expf(v1 - m); v2 = expf(v2 - m); v3 = expf(v3 - m);
  red[t] = v0 + v1 + v2 + v3;
  __syncthreads();
  for (int s = 128; s > 0; s >>= 1) {
    if (t < s) red[t] += red[t + s];
    __syncthreads();
  }
  const float inv = 1.f / red[0];
  out[t] = v0 * inv; out[t + 256] = v1 * inv; out[t + 512] = v2 * inv; out[t + 768] = v3 * inv;
}

// ---------------------------------------------------------------------------
// Host-side orchestration.
// d_in order: 0:input, 1..10:(h1,c1..h5,c5), 11..30:(Wih,Whh,bih,bhh)x5,
//             31:Wout, 32:bout
// d_out: [out(1024) | hn1 cn1 hn2 cn2 hn3 cn3 hn4 cn4 hn5 cn5] (2048 each)
// ---------------------------------------------------------------------------
extern "C" void kernel_launch(void* const* d_in, const int* in_sizes, int n_in,
                              void* d_out, int out_size, void* d_ws, size_t ws_size,
                              hipStream_t stream) {
  (void)in_sizes; (void)n_in; (void)out_size; (void)ws_size;

  const float* input = (const float*)d_in[0];
  const float *h[5], *c[5], *Wih[5], *Whh[5], *bih[5], *bhh[5];
  for (int k = 0; k < 5; ++k) {
    h[k]   = (const float*)d_in[1 + 2 * k];
    c[k]   = (const float*)d_in[2 + 2 * k];
    Wih[k] = (const float*)d_in[11 + 4 * k];
    Whh[k] = (const float*)d_in[12 + 4 * k];
    bih[k] = (const float*)d_in[13 + 4 * k];
    bhh[k] = (const float*)d_in[14 + 4 * k];
  }
  const float* Wout = (const float*)d_in[31];
  const float* bout = (const float*)d_in[32];

  float* out = (float*)d_out;
  float* wsf = (float*)d_ws;
  float* z    = wsf;                 // 5 * 8192
  float* g    = wsf + 5 * GSZ;       // 8192 (reused per layer, serial chain)
  float* ybuf = g + GSZ;             // 5 * 1024
  float* xbuf = ybuf + 5 * OSZ;      // 4 * 1024 (inputs for layers 2..5)

  // Stage 1: all recurrent GEMVs (independent of the chain) in one wide kernel.
  Whh5Args wa;
  for (int k = 0; k < 5; ++k) {
    wa.W[k] = Whh[k]; wa.h[k] = h[k]; wa.b0[k] = bih[k]; wa.b1[k] = bhh[k];
  }
  wa.z = z;
  k_whh5<<<320, 256, 0, stream>>>(wa);

  // Stage 2: the serial layer chain.
  for (int k = 0; k < 5; ++k) {
    const float* xk = (k == 0) ? input : (xbuf + (k - 1) * OSZ);
    k_gemv_add<<<64, 256, 0, stream>>>(Wih[k], xk, z + k * GSZ, g, OSZ);

    float* hn = out + OSZ + k * (2 * HSZ);
    float* cn = hn + HSZ;
    k_lstm_pointwise<<<8, 256, 0, stream>>>(g, c[k], hn, cn);

    float* yk = ybuf + k * OSZ;
    const float* yprev = (k >= 1) ? (ybuf + (k - 1) * OSZ) : nullptr;
    float* xnext = (k < 4) ? (xbuf + k * OSZ) : nullptr;
    k_head<<<8, 256, 0, stream>>>(Wout, hn, bout, yk, yprev, xnext);
  }

  // Stage 3: softmax of the final head output.
  k_softmax<<<1, 256, 0, stream>>>(ybuf + 4 * OSZ, out);
}